// Seq2SeqSelfAttention_61847529062614
// MI455X (gfx1250) — compile-verified
//
#include <hip/hip_runtime.h>
#include <hip/hip_bf16.h>

#define B_  16
#define H_  16
#define NB_ 32
#define BS_ 128
#define DH_ 64
#define DM_ 1024
#define KV_ (NB_*BS_)
#define ROWF_ 68              // padded K-tile row stride (floats): banks step by 4

typedef __attribute__((ext_vector_type(2))) float v2f;
typedef __attribute__((ext_vector_type(4))) float v4f;
typedef __attribute__((ext_vector_type(8))) float v8f;
typedef __attribute__((ext_vector_type(4))) int   v4i;

// ---------------------------------------------------------------------------
// CDNA5 async global->LDS copy path (ASYNCcnt), with clean fallback.
// Builtin confirmed present; signature: (int4 AS1*, int4 AS3*, imm, imm).
// ---------------------------------------------------------------------------
#if __has_builtin(__builtin_amdgcn_global_load_async_to_lds_b128)
#define HAVE_ASYNC 1
__device__ __forceinline__ void async_cp16(void* lds, const void* g) {
    __builtin_amdgcn_global_load_async_to_lds_b128(
        (__attribute__((address_space(1))) v4i*)g,
        (__attribute__((address_space(3))) v4i*)lds, 0, 0);
}
__device__ __forceinline__ void async_wait_le8() {
#if __has_builtin(__builtin_amdgcn_s_wait_asynccnt)
    __builtin_amdgcn_s_wait_asynccnt(8);
#else
    asm volatile("s_wait_asynccnt 0x8" ::: "memory");
#endif
}
__device__ __forceinline__ void async_wait_0() {
#if __has_builtin(__builtin_amdgcn_s_wait_asynccnt)
    __builtin_amdgcn_s_wait_asynccnt(0);
#else
    asm volatile("s_wait_asynccnt 0x0" ::: "memory");
#endif
}
#else
#define HAVE_ASYNC 0
__device__ __forceinline__ void async_cp16(void* lds, const void* g) {
    *(v4f*)lds = *(const v4f*)g;     // synchronous staging fallback
}
__device__ __forceinline__ void async_wait_le8() {}
__device__ __forceinline__ void async_wait_0()   {}
#endif

// ---------------------------------------------------------------------------
// One wave (32 threads) computes Y[0:16, n0:n0+16] = X(16xK) @ W(NxK)^T + bias
// using V_WMMA_F32_16X16X4_F32 (exact f32 semantics, matches reference).
// ---------------------------------------------------------------------------
__device__ __forceinline__ void gemm16_tile_f32wmma(
    const float* __restrict__ X, const float* __restrict__ W,
    const float* __restrict__ bias, float* __restrict__ Y, int n0)
{
    __shared__ float sA[16 * 64];
    __shared__ float sB[16 * 64];
    const int lane = threadIdx.x;     // 0..31
    const int mr   = lane & 15;
    const int half = lane >> 4;

    v8f acc = {};
    for (int kk0 = 0; kk0 < DM_; kk0 += 64) {
        #pragma unroll
        for (int r = 0; r < 8; ++r) {
            int j = lane + 32 * r;
            int m = j >> 4;
            int c = (j & 15) << 2;
            *(v4f*)&sA[m * 64 + c] = *(const v4f*)&X[(size_t)m * DM_ + kk0 + c];
            *(v4f*)&sB[m * 64 + c] = *(const v4f*)&W[(size_t)(n0 + m) * DM_ + kk0 + c];
        }
        __syncthreads();
        #pragma unroll
        for (int k4 = 0; k4 < 64; k4 += 4) {
            v2f a, b;
            a.x = sA[mr * 64 + k4 + 2 * half];
            a.y = sA[mr * 64 + k4 + 2 * half + 1];
            b.x = sB[mr * 64 + k4 + 2 * half];
            b.y = sB[mr * 64 + k4 + 2 * half + 1];
            acc = __builtin_amdgcn_wmma_f32_16x16x4_f32(
                false, a, false, b, (short)0, acc, false, false);
        }
        __syncthreads();
    }
    #pragma unroll
    for (int j = 0; j < 8; ++j) {
        int m = j + 8 * half;
        int n = n0 + mr;
        Y[(size_t)m * DM_ + n] = acc[j] + bias[n];
    }
}

__global__ __launch_bounds__(32) void qkv_proj_kernel(
    const float* __restrict__ hs,
    const float* __restrict__ Wq, const float* __restrict__ bq,
    const float* __restrict__ Wk, const float* __restrict__ bk,
    const float* __restrict__ Wv, const float* __restrict__ bv,
    float* __restrict__ q, float* __restrict__ k, float* __restrict__ v)
{
    int n0    = (blockIdx.x & 63) * 16;
    int which = blockIdx.x >> 6;
    if (which == 0)       gemm16_tile_f32wmma(hs, Wq, bq, q, n0);
    else if (which == 1)  gemm16_tile_f32wmma(hs, Wk, bk, k, n0);
    else                  gemm16_tile_f32wmma(hs, Wv, bv, v, n0);
}

__global__ __launch_bounds__(32) void out_proj_kernel(
    const float* __restrict__ ctx, const float* __restrict__ Wo,
    const float* __restrict__ bo, float* __restrict__ out)
{
    gemm16_tile_f32wmma(ctx, Wo, bo, out, blockIdx.x * 16);
}

// ---------------------------------------------------------------------------
// Attention: one block per (b, h). K streamed via double-buffered async
// global->LDS tiles (one 32KB physical page per tile); V streamed with
// non-temporal coalesced loads. Cache update at `pos` is virtualized.
// ---------------------------------------------------------------------------
__device__ __forceinline__ void issue_k_tile(
    const float* __restrict__ k_cache, const int* __restrict__ block_tables,
    int b, int h, int t, float* __restrict__ kbuf, int tid)
{
    int pb = block_tables[b * NB_ + t];
    const float* src =
        k_cache + (((size_t)b * H_ + h) * NB_ + pb) * (size_t)(BS_ * DH_);
    #pragma unroll
    for (int r = 0; r < 8; ++r) {
        int j   = tid + 256 * r;   // 0..2047 : 16B chunk id, coalesced
        int row = j >> 4;          // 0..127
        int ch  = j & 15;          // chunk within row
        async_cp16(&kbuf[row * ROWF_ + ch * 4], src + row * DH_ + ch * 4);
    }
}

__global__ __launch_bounds__(256) void attn_kernel(
    const float* __restrict__ q, const float* __restrict__ k_new,
    const float* __restrict__ v_new,
    const float* __restrict__ k_cache, const float* __restrict__ v_cache,
    const float* __restrict__ mask, const int* __restrict__ cache_position,
    const int* __restrict__ block_tables, float* __restrict__ ctx_out)
{
    extern __shared__ float smem[];
    float* s_scores = smem;                  // KV_ floats
    float* s_q      = smem + KV_;            // DH_
    float* s_red    = s_q + DH_;             // 256
    float* s_kbuf   = s_red + 256;           // 2 * 128 * ROWF_

    const int tid = threadIdx.x;
    const int bh  = blockIdx.x;
    const int b   = bh / H_;
    const int h   = bh % H_;
    const int pos = cache_position[b];
    const int ntiles = (pos >> 7) + 1;       // causal: only tiles touching [0,pos]

    // prologue: start tile 0 while we initialize scores and load q
    issue_k_tile(k_cache, block_tables, b, h, 0, s_kbuf, tid);

    if (tid < DH_) s_q[tid] = q[(size_t)b * DM_ + h * DH_ + tid];
    for (int kv = tid; kv < KV_; kv += 256) s_scores[kv] = -1.0e9f;
    __syncthreads();

    // ---- phase 1: scores, double-buffered K tiles ----
    const int row  = tid & 127;
    const int half = tid >> 7;               // 2 threads per row, 32 dims each
    for (int t = 0; t < ntiles; ++t) {
        float* kb = s_kbuf + (t & 1) * (128 * ROWF_);
        if (t + 1 < ntiles) {
            issue_k_tile(k_cache, block_tables, b, h, t + 1,
                         s_kbuf + ((t + 1) & 1) * (128 * ROWF_), tid);
            async_wait_le8();                // retire tile t, keep t+1 in flight
        } else {
            async_wait_0();
        }
        __syncthreads();                     // tile t fully resident in LDS

        int kv = t * BS_ + row;
        float part = 0.f;
        if (kv <= pos) {
            const float* kr = (kv == pos)
                ? (k_new + (size_t)b * DM_ + h * DH_ + 32 * half)
                : (kb + row * ROWF_ + 32 * half);
            const float* qh = s_q + 32 * half;
            #pragma unroll
            for (int i = 0; i < 32; ++i) part += kr[i] * qh[i];
        }
        s_red[tid] = part;
        __syncthreads();
        if (tid < 128) {
            int kvr = t * BS_ + tid;
            if (kvr <= pos)
                s_scores[kvr] = s_red[tid] + s_red[tid + 128] +
                    __builtin_nontemporal_load(mask + (size_t)bh * KV_ + kvr);
        }
        __syncthreads();                     // reads done before buffer reuse
    }

    // ---- softmax: block max ----
    float lmax = -1.0e30f;
    for (int kv = tid; kv < KV_; kv += 256) lmax = fmaxf(lmax, s_scores[kv]);
    s_red[tid] = lmax;
    __syncthreads();
    for (int st = 128; st > 0; st >>= 1) {
        if (tid < st) s_red[tid] = fmaxf(s_red[tid], s_red[tid + st]);
        __syncthreads();
    }
    const float gmax = s_red[0];
    __syncthreads();

    // ---- exp + block sum ----
    float lsum = 0.f;
    for (int kv = tid; kv < KV_; kv += 256) {
        float e = __expf(s_scores[kv] - gmax);
        s_scores[kv] = e;
        lsum += e;
    }
    s_red[tid] = lsum;
    __syncthreads();
    for (int st = 128; st > 0; st >>= 1) {
        if (tid < st) s_red[tid] += s_red[tid + st];
        __syncthreads();
    }
    const float inv = 1.0f / s_red[0];
    __syncthreads();

    // ---- ctx = probs @ V : 4 kv-chunks x 64 dims, NT coalesced row reads ----
    const int d     = tid & 63;
    const int chunk = tid >> 6;
    float partial = 0.f;
    const int kv_lo = chunk * (KV_ / 4);
    const int kv_hi = kv_lo + (KV_ / 4);
    for (int kv = kv_lo; kv < kv_hi; ++kv) {
        if (kv > pos) break;                 // probs exactly zero beyond pos
        float p = s_scores[kv];
        const float* vrow;
        if (kv == pos) {
            vrow = v_new + (size_t)b * DM_ + h * DH_;
        } else {
            int pb = block_tables[b * NB_ + (kv >> 7)];
            vrow = v_cache +
                ((((size_t)b * H_ + h) * NB_ + pb) * BS_ + (kv & 127)) * DH_;
        }
        partial += p * __builtin_nontemporal_load(vrow + d);
    }
    s_red[tid] = partial;
    __syncthreads();
    if (tid < DH_) {
        float c = (s_red[tid] + s_red[64 + tid] + s_red[128 + tid] +
                   s_red[192 + tid]) * inv;
        ctx_out[(size_t)b * DM_ + h * DH_ + tid] = c;
    }
}

extern "C" void kernel_launch(void* const* d_in, const int* in_sizes, int n_in,
                              void* d_out, int out_size, void* d_ws, size_t ws_size,
                              hipStream_t stream) {
    const float* hs   = (const float*)d_in[0];
    const float* kc   = (const float*)d_in[1];
    const float* vc   = (const float*)d_in[2];
    const float* mask = (const float*)d_in[3];
    const int*   cpos = (const int*)d_in[4];
    const int*   btab = (const int*)d_in[5];
    const float* Wq = (const float*)d_in[6];  const float* bq = (const float*)d_in[7];
    const float* Wk = (const float*)d_in[8];  const float* bk = (const float*)d_in[9];
    const float* Wv = (const float*)d_in[10]; const float* bv = (const float*)d_in[11];
    const float* Wo = (const float*)d_in[12]; const float* bo = (const float*)d_in[13];

    float* ws     = (float*)d_ws;
    float* q_ws   = ws;                 // 16x1024
    float* k_ws   = ws + 1 * B_ * DM_;  // 16x1024
    float* v_ws   = ws + 2 * B_ * DM_;  // 16x1024
    float* ctx_ws = ws + 3 * B_ * DM_;  // 16x1024

    const size_t shmem =
        (size_t)(KV_ + DH_ + 256 + 2 * 128 * ROWF_) * sizeof(float); // ~87 KB

    qkv_proj_kernel<<<192, 32, 0, stream>>>(hs, Wq, bq, Wk, bk, Wv, bv,
                                            q_ws, k_ws, v_ws);
    attn_kernel<<<B_ * H_, 256, shmem, stream>>>(q_ws, k_ws, v_ws, kc, vc, mask,
                                                 cpos, btab, ctx_ws);
    out_proj_kernel<<<64, 32, 0, stream>>>(ctx_ws, Wo, bo, (float*)d_out);
}